// PriorEncoder_36481452212629
// MI455X (gfx1250) — compile-verified
//
#include <hip/hip_runtime.h>
#include <hip/hip_bf16.h>

typedef __attribute__((ext_vector_type(2))) float v2f;
typedef __attribute__((ext_vector_type(8))) float v8f;

#define BSZ 8
#define CCH 192
#define TX  512
#define TY  2048
#define NEG_INF_F (-1e9f)

// ---------------------------------------------------------------------------
// Kernel 1: build B' = [s ; m*s] (K-major, [b][2C][Tx]) and nc14[b][x]
// ---------------------------------------------------------------------------
__global__ void __launch_bounds__(256)
prep_kernel(const float* __restrict__ logs_p, const float* __restrict__ m_p,
            float* __restrict__ Bp, float* __restrict__ nc14) {
    int gid = blockIdx.x * 256 + threadIdx.x;       // 0..4095
    int b = gid >> 9;
    int x = gid & (TX - 1);
    const float c1 = -0.5f * 1.8378770664093453f;   // -0.5*log(2*pi)
    float suml = 0.f, snc4 = 0.f;
    int base = b * CCH * TX + x;
    int obase = b * 2 * CCH * TX + x;
    for (int c = 0; c < CCH; ++c) {
        float l = logs_p[base + c * TX];
        float m = m_p[base + c * TX];
        float s = expf(-2.f * l);
        Bp[obase + c * TX] = s;
        Bp[obase + (CCH + c) * TX] = m * s;
        suml += l;
        snc4 += m * m * s;
    }
    nc14[b * TX + x] = (float)CCH * c1 - suml - 0.5f * snc4;
}

// ---------------------------------------------------------------------------
// Kernel 2: neg_cent GEMM via V_WMMA_F32_16X16X4_F32, one wave per 16x16 tile
//   D[t,x] = sum_c (-0.5*z^2)[t,c]*s[c,x] + z[t,c]*(m*s)[c,x]  + nc14[x]
// A fragment per lane: (row M = lane%16, K = k0 + 2*(lane>=16) + {0,1})
// B fragment per lane: (col N = lane%16, same K pair) -- mirrored layout
// D: VGPR j -> row M = 16*tm + j + 8*(lane>=16), col N = 16*tn + lane%16
// ---------------------------------------------------------------------------
__global__ void __launch_bounds__(256)
negcent_kernel(const float* __restrict__ zp, const float* __restrict__ Bp,
               const float* __restrict__ nc14, float* __restrict__ negc) {
    int wave = blockIdx.x * 8 + (threadIdx.x >> 5);  // 0..32767
    int lane = threadIdx.x & 31;
    int tn = wave & 31;
    int tm = (wave >> 5) & 127;
    int b  = wave >> 12;
    int hi  = lane >> 4;       // 0 or 1
    int m16 = lane & 15;
    int t = tm * 16 + m16;
    int x = tn * 16 + m16;
    const float* zb = zp + (size_t)b * CCH * TY + t;       // index + c*TY
    const float* bb = Bp + (size_t)b * 2 * CCH * TX + x;   // index + k*TX
    v8f acc = {};
    // part 1: A = -0.5*z^2, B = s
    #pragma unroll 4
    for (int k0 = 0; k0 < CCH; k0 += 4) {
        int c = k0 + 2 * hi;
        float z0 = zb[c * TY];
        float z1 = zb[(c + 1) * TY];
        v2f a;  a.x = -0.5f * z0 * z0;  a.y = -0.5f * z1 * z1;
        v2f bv; bv.x = bb[c * TX];      bv.y = bb[(c + 1) * TX];
        acc = __builtin_amdgcn_wmma_f32_16x16x4_f32(false, a, false, bv,
                                                    (short)0, acc, false, false);
    }
    // part 2: A = z, B = m*s
    #pragma unroll 4
    for (int k0 = 0; k0 < CCH; k0 += 4) {
        int c = k0 + 2 * hi;
        float z0 = zb[c * TY];
        float z1 = zb[(c + 1) * TY];
        v2f a;  a.x = z0;  a.y = z1;
        v2f bv; bv.x = bb[(CCH + c) * TX];  bv.y = bb[(CCH + c + 1) * TX];
        acc = __builtin_amdgcn_wmma_f32_16x16x4_f32(false, a, false, bv,
                                                    (short)0, acc, false, false);
    }
    float addl = nc14[b * TX + x];
    int rbase = tm * 16 + hi * 8;
    #pragma unroll
    for (int j = 0; j < 8; ++j) {
        negc[((size_t)b * TY + rbase + j) * TX + x] = acc[j] + addl;
    }
}

// ---------------------------------------------------------------------------
// Kernel 3: MAS forward DP. One block (512 threads) per batch; y-serial.
// dirs bitmask: word (b*TY + y)*16 + x/32, bit x%32 = (value[y-1][x-1] > value[y-1][x])
// ---------------------------------------------------------------------------
__global__ void __launch_bounds__(512)
mas_forward(const float* __restrict__ negc, unsigned* __restrict__ dirs) {
    int b = blockIdx.x;
    int x = threadIdx.x;
    __shared__ float lds[TX];
    float v = NEG_INF_F;
    const float* row = negc + (size_t)b * TY * TX + x;
    unsigned* drow = dirs + (size_t)b * TY * 16 + (x >> 5);
    for (int y = 0; y < TY; ++y) {
        float score = row[(size_t)y * TX];
        __syncthreads();
        lds[x] = v;
        __syncthreads();
        float diag = (x == 0) ? NEG_INF_F : lds[x - 1];
        bool d = diag > v;
        unsigned bal = (unsigned)__ballot(d);   // wave32: low 32 bits
        if ((x & 31) == 0) drow[(size_t)y * 16] = bal;
        float nv = score + fmaxf(v, diag);
        if (y == 0) nv = (x == 0) ? score : NEG_INF_F;
        v = nv;
    }
}

// ---------------------------------------------------------------------------
// Kernel 4: backtrack (one block per batch). LDS-staged dirs in 512-row
// chunks (y descending). Produces idxbuf (x index per y, -1 inactive),
// duration (d_out) and spec_len (for loss denominator).
// ---------------------------------------------------------------------------
__global__ void __launch_bounds__(256)
mas_backtrack(const unsigned* __restrict__ dirs,
              const float* __restrict__ text_mask, const float* __restrict__ spec_mask,
              int* __restrict__ idxbuf, float* __restrict__ duration_out,
              float* __restrict__ spec_len_out) {
    int b = blockIdx.x;
    int tid = threadIdx.x;
    __shared__ unsigned ldsDirs[512 * 16];   // 32 KB
    __shared__ int      ldsIdx[512];
    __shared__ unsigned ldsDur[TX];
    __shared__ float    red[256];
    __shared__ int      sLens[2];

    // text_len
    float tsum = 0.f;
    for (int x = tid; x < TX; x += 256) tsum += text_mask[b * TX + x];
    red[tid] = tsum; __syncthreads();
    for (int s = 128; s > 0; s >>= 1) { if (tid < s) red[tid] += red[tid + s]; __syncthreads(); }
    if (tid == 0) sLens[0] = (int)(red[0] + 0.5f);
    __syncthreads();
    // spec_len
    float ssum = 0.f;
    for (int y = tid; y < TY; y += 256) ssum += spec_mask[b * TY + y];
    red[tid] = ssum; __syncthreads();
    for (int s = 128; s > 0; s >>= 1) { if (tid < s) red[tid] += red[tid + s]; __syncthreads(); }
    if (tid == 0) { sLens[1] = (int)(red[0] + 0.5f); spec_len_out[b] = red[0]; }
    for (int x = tid; x < TX; x += 256) ldsDur[x] = 0u;
    __syncthreads();

    int text_len = sLens[0];
    int spec_len = sLens[1];
    int idx = text_len - 1;                       // live only in thread 0

    for (int chunk = 3; chunk >= 0; --chunk) {
        int ybase = chunk * 512;
        for (int i = tid; i < 512 * 16; i += 256)
            ldsDirs[i] = dirs[(size_t)b * TY * 16 + (size_t)ybase * 16 + i];
        __syncthreads();
        if (tid == 0) {
            for (int yy = 511; yy >= 0; --yy) {
                int y = ybase + yy;
                bool active = y < spec_len;
                ldsIdx[yy] = active ? idx : -1;
                unsigned w = ldsDirs[yy * 16 + (idx >> 5)];
                bool dat = (w >> (idx & 31)) & 1u;
                bool move = (idx != 0) && ((idx == y) || dat);
                if (active && move) idx--;
            }
        }
        __syncthreads();
        for (int yy = tid; yy < 512; yy += 256) {
            int vi = ldsIdx[yy];
            idxbuf[b * TY + ybase + yy] = vi;
            if (vi >= 0) atomicAdd(&ldsDur[vi], 1u);
        }
        __syncthreads();
    }
    for (int x = tid; x < TX; x += 256)
        duration_out[b * TX + x] = (float)ldsDur[x];
}

// ---------------------------------------------------------------------------
// Kernel 5: write hard path (overwrites the neg_cent region of d_out)
// ---------------------------------------------------------------------------
__global__ void __launch_bounds__(256)
write_path(const int* __restrict__ idxbuf, float* __restrict__ path) {
    int gid = blockIdx.x;                 // b*TY + y
    int idx = idxbuf[gid];
    float* row = path + (size_t)gid * TX;
    for (int x = threadIdx.x; x < TX; x += 256)
        row[x] = (x == idx) ? 1.f : 0.f;
}

// ---------------------------------------------------------------------------
// Kernel 6: gather m_p_a / logs_p_a and KL partial sums (one block per (b,c))
// ---------------------------------------------------------------------------
__global__ void __launch_bounds__(256)
gather_kl(const int* __restrict__ idxbuf, const float* __restrict__ m_p,
          const float* __restrict__ logs_p, const float* __restrict__ z_p,
          const float* __restrict__ logs_q,
          float* __restrict__ m_p_a, float* __restrict__ logs_p_a,
          float* __restrict__ partials) {
    int bc = blockIdx.x;                  // 0..B*C-1
    int b = bc / CCH;
    const float* mrow = m_p    + (size_t)bc * TX;
    const float* lrow = logs_p + (size_t)bc * TX;
    const float* zrow = z_p    + (size_t)bc * TY;
    const float* qrow = logs_q + (size_t)bc * TY;
    float* mo = m_p_a    + (size_t)bc * TY;
    float* lo = logs_p_a + (size_t)bc * TY;
    const int* ib = idxbuf + b * TY;
    float acc = 0.f;
    for (int y = threadIdx.x; y < TY; y += 256) {
        int idx = ib[y];
        bool act = idx >= 0;
        float mv = act ? mrow[idx] : 0.f;
        float lv = act ? lrow[idx] : 0.f;
        mo[y] = mv;
        lo[y] = lv;
        if (act) {
            float dz = zrow[y] - mv;
            acc += lv - qrow[y] - 0.5f + 0.5f * dz * dz * expf(-2.f * lv);
        }
    }
    __shared__ float red[256];
    red[threadIdx.x] = acc; __syncthreads();
    for (int s = 128; s > 0; s >>= 1) {
        if (threadIdx.x < s) red[threadIdx.x] += red[threadIdx.x + s];
        __syncthreads();
    }
    if (threadIdx.x == 0) partials[bc] = red[0];
}

// ---------------------------------------------------------------------------
// Kernel 7: deterministic loss finalize
// ---------------------------------------------------------------------------
__global__ void finalize_loss(const float* __restrict__ partials,
                              const float* __restrict__ spec_len,
                              float* __restrict__ loss) {
    if (threadIdx.x == 0 && blockIdx.x == 0) {
        float num = 0.f;
        for (int i = 0; i < BSZ * CCH; ++i) num += partials[i];
        float den = 0.f;
        for (int i = 0; i < BSZ; ++i) den += spec_len[i];
        *loss = num / den;
    }
}

// ---------------------------------------------------------------------------
extern "C" void kernel_launch(void* const* d_in, const int* in_sizes, int n_in,
                              void* d_out, int out_size, void* d_ws, size_t ws_size,
                              hipStream_t stream) {
    const float* z_p       = (const float*)d_in[0];   // [8,192,2048]
    const float* m_p       = (const float*)d_in[1];   // [8,192,512]
    const float* logs_p    = (const float*)d_in[2];   // [8,192,512]
    const float* logs_q    = (const float*)d_in[3];   // [8,192,2048]
    const float* text_mask = (const float*)d_in[4];   // [8,1,512]
    const float* spec_mask = (const float*)d_in[5];   // [8,1,2048]

    float* out = (float*)d_out;
    // output layout (floats)
    float* path      = out;                                   // 8*2048*512
    float* m_p_a     = out + (size_t)BSZ * TY * TX;           // 8*192*2048
    float* logs_p_a  = m_p_a + (size_t)BSZ * CCH * TY;
    float* loss      = logs_p_a + (size_t)BSZ * CCH * TY;     // 1
    float* duration  = loss + 1;                              // 8*512

    // neg_cent temporarily lives in the path region (same element count)
    float* negc = path;

    // workspace layout
    char* ws = (char*)d_ws;
    float*    Bp       = (float*)ws;                               // 8*384*512
    float*    nc14     = Bp + (size_t)BSZ * 2 * CCH * TX;          // 8*512
    unsigned* dirs     = (unsigned*)(nc14 + BSZ * TX);             // 8*2048*16
    int*      idxbuf   = (int*)(dirs + (size_t)BSZ * TY * 16);     // 8*2048
    float*    spec_len = (float*)(idxbuf + BSZ * TY);              // 8
    float*    partials = spec_len + BSZ;                           // 8*192

    // 1. B' and bias
    prep_kernel<<<dim3((BSZ * TX) / 256), dim3(256), 0, stream>>>(logs_p, m_p, Bp, nc14);
    // 2. neg_cent GEMM (WMMA f32 16x16x4), 32768 tiles / 8 waves per block
    negcent_kernel<<<dim3(4096), dim3(256), 0, stream>>>(z_p, Bp, nc14, negc);
    // 3. MAS forward DP
    mas_forward<<<dim3(BSZ), dim3(512), 0, stream>>>(negc, dirs);
    // 4. backtrack + duration + lengths
    mas_backtrack<<<dim3(BSZ), dim3(256), 0, stream>>>(dirs, text_mask, spec_mask,
                                                       idxbuf, duration, spec_len);
    // 5. hard path (overwrites neg_cent region)
    write_path<<<dim3(BSZ * TY), dim3(256), 0, stream>>>(idxbuf, path);
    // 6. gathers + KL partials
    gather_kl<<<dim3(BSZ * CCH), dim3(256), 0, stream>>>(idxbuf, m_p, logs_p, z_p, logs_q,
                                                         m_p_a, logs_p_a, partials);
    // 7. deterministic loss
    finalize_loss<<<dim3(1), dim3(32), 0, stream>>>(partials, spec_len, loss);
}